// EMAUpdater_8409545966131
// MI455X (gfx1250) — compile-verified
//
#include <hip/hip_runtime.h>
#include <hip/hip_bf16.h>
#include <stdint.h>

// Problem constants (from reference)
#define BATCH   16384
#define CODEDIM 256
#define BOOK    8192
#define GAMMA   0.99f
#define ONEMG   0.01f   // 1 - GAMMA

// Tiling
#define CDIM  4                  // dims owned per workgroup
#define NWG   (CODEDIM / CDIM)   // 64 workgroups, one full-K bin slice each
#define TPB   256                // 8 wave32s
#define ITERS (BATCH / TPB)      // 64 double-buffered iterations

// LDS layout (bytes) — exploits CDNA5's 320KB-per-workgroup LDS
#define BINS_OFF     0
#define BINS_BYTES   (BOOK * CDIM * 4)            // 131072: float bins[8192][4]
#define CNT_OFF      (BINS_OFF + BINS_BYTES)
#define CNT_BYTES    (BOOK * 4)                   // 32768: uint cnt[8192] (WG0 only)
#define STG_IN_OFF   (CNT_OFF + CNT_BYTES)        // 163840
#define STG_IN_BYTES (2 * TPB * 16)               // 8192: double-buffered float4 stage
#define STG_IX_OFF   (STG_IN_OFF + STG_IN_BYTES)  // 172032
#define STG_IX_BYTES (2 * TPB * 4)                // 2048: double-buffered idx stage
#define LDS_TOTAL    (STG_IX_OFF + STG_IX_BYTES)  // 174080 bytes < 320KB

extern __shared__ char smem[];

// Kernel 1: segment-sum via LDS binning.
// WG w owns dims [w*4, w*4+4) for ALL 8192 codes. Streams all 16384 rows
// through async-to-LDS staging (per-lane gather of the strided 16B column
// slice), then ds_add_f32 into the LDS bin slice. WG0 also histograms idx
// and emits N_new. No global atomics anywhere.
__global__ void ema_scatter(const float* __restrict__ inputs,
                            const int* __restrict__ idx,
                            const float* __restrict__ N_in,
                            float* __restrict__ out) {
  float*    bins     = (float*)(smem + BINS_OFF);
  unsigned* cnt      = (unsigned*)(smem + CNT_OFF);
  float4*   stage_in = (float4*)(smem + STG_IN_OFF);
  int*      stage_ix = (int*)(smem + STG_IX_OFF);

  const int w = blockIdx.x;   // dim-slice id
  const int t = threadIdx.x;

  for (int i = t; i < BOOK * CDIM; i += TPB) bins[i] = 0.0f;
  if (w == 0) {
    for (int i = t; i < BOOK; i += TPB) cnt[i] = 0u;
  }
  __syncthreads();

  // Per-lane LDS byte addresses of this lane's private staging slots.
  // (Low 32 bits of a flat shared pointer == LDS byte address.)
  const uint32_t lds_base = (uint32_t)(uintptr_t)smem;
  const uint32_t in_a[2] = { lds_base + STG_IN_OFF + (uint32_t)t * 16u,
                             lds_base + STG_IN_OFF + (uint32_t)(TPB + t) * 16u };
  const uint32_t ix_a[2] = { lds_base + STG_IX_OFF + (uint32_t)t * 4u,
                             lds_base + STG_IX_OFF + (uint32_t)(TPB + t) * 4u };

  // Prefetch iteration 0 into buffer 0 (2 async ops per wave).
  {
    const float* g  = inputs + ((size_t)t * CODEDIM + (size_t)w * CDIM);
    const int*   gi = idx + t;
    asm volatile("global_load_async_to_lds_b128 %0, %1, off"
                 :: "v"(in_a[0]), "v"(g) : "memory");
    asm volatile("global_load_async_to_lds_b32 %0, %1, off"
                 :: "v"(ix_a[0]), "v"(gi) : "memory");
  }

  for (int it = 0; it < ITERS; ++it) {
    const int buf = it & 1;
    if (it + 1 < ITERS) {
      // Issue next tile's DMA, then wait only for the current tile
      // (async loads complete in order; 2 newest may stay in flight).
      const int    nb  = buf ^ 1;
      const size_t row = (size_t)(it + 1) * TPB + (size_t)t;
      const float* g   = inputs + (row * CODEDIM + (size_t)w * CDIM);
      const int*   gi  = idx + row;
      asm volatile("global_load_async_to_lds_b128 %0, %1, off"
                   :: "v"(in_a[nb]), "v"(g) : "memory");
      asm volatile("global_load_async_to_lds_b32 %0, %1, off"
                   :: "v"(ix_a[nb]), "v"(gi) : "memory");
      asm volatile("s_wait_asynccnt 2" ::: "memory");
    } else {
      asm volatile("s_wait_asynccnt 0" ::: "memory");
    }

    // Each lane consumes only the slot it DMA'd => no barrier needed.
    const float4 v   = stage_in[buf * TPB + t];
    const int    bin = stage_ix[buf * TPB + t];
    float* bp = bins + (size_t)bin * CDIM;
    atomicAdd(bp + 0, v.x);   // ds_add_f32
    atomicAdd(bp + 1, v.y);
    atomicAdd(bp + 2, v.z);
    atomicAdd(bp + 3, v.w);
    if (w == 0) atomicAdd(&cnt[bin], 1u);  // ds_add_u32
  }

  __syncthreads();

  // d_out layout: [codebook_new K*D][N_new K][m_new K*D]
  float* out_N = out + (size_t)BOOK * CODEDIM;
  float* out_m = out_N + BOOK;

  // Spill raw sums into the m_new region (every (k,dim) covered exactly once
  // across the 64 WGs, so no pre-zeroing of d_out is needed).
  for (int k = t; k < BOOK; k += TPB) {
    const float4 s = *(const float4*)(bins + (size_t)k * CDIM);
    *(float4*)(out_m + (size_t)k * CODEDIM + (size_t)w * CDIM) = s;
  }
  if (w == 0) {
    for (int k = t; k < BOOK; k += TPB) {
      out_N[k] = GAMMA * N_in[k] + ONEMG * (float)cnt[k];  // N_new
    }
  }
}

// Kernel 2: m_new = gamma*m + (1-gamma)*sums (in place over the raw sums),
// codebook_new = m_new / N_new. One thread per (k, d); no races.
__global__ void ema_finalize(const float* __restrict__ m_in,
                             float* __restrict__ out) {
  float* out_cb = out;
  float* out_N  = out + (size_t)BOOK * CODEDIM;
  float* out_m  = out_N + BOOK;

  const int    k = blockIdx.x;
  const int    d = threadIdx.x;
  const size_t e = (size_t)k * CODEDIM + (size_t)d;

  const float Nn = out_N[k];                               // written by kernel 1
  const float mn = GAMMA * m_in[e] + ONEMG * out_m[e];     // out_m holds raw sums
  out_m[e]  = mn;
  out_cb[e] = mn / Nn;
}

extern "C" void kernel_launch(void* const* d_in, const int* in_sizes, int n_in,
                              void* d_out, int out_size, void* d_ws, size_t ws_size,
                              hipStream_t stream) {
  const float* inputs = (const float*)d_in[0];
  // d_in[1] = distances: dead in the reference body — never read (saves 512MB).
  const int*   idx    = (const int*)d_in[2];   // JAX default x64-off => int32
  // d_in[3] = codebook values: unused (only its shape matters).
  const float* N_in   = (const float*)d_in[4];
  const float* m_in   = (const float*)d_in[5];
  float* out = (float*)d_out;
  (void)in_sizes; (void)n_in; (void)out_size; (void)d_ws; (void)ws_size;

  ema_scatter<<<NWG, TPB, LDS_TOTAL, stream>>>(inputs, idx, N_in, out);
  ema_finalize<<<BOOK, CODEDIM, 0, stream>>>(m_in, out);
}